// DGCFP_14027363188882
// MI455X (gfx1250) — compile-verified
//
#include <hip/hip_runtime.h>

typedef float v2f __attribute__((ext_vector_type(2)));
typedef float v8f __attribute__((ext_vector_type(8)));

#define B_      8
#define N_      4096
#define NPOINT_ 1024
#define NSAMPLE_ 32
#define CCH_    64          // point feature channels
#define OCH_    67          // 3 + 64
#define R2_     (0.2f * 0.2f)

// ---------------------------------------------------------------------------
// Kernel A: farthest point sampling. 1 block per batch, 1024 threads,
// each thread owns points t, t+1024, t+2048, t+3072 in registers.
// Writes new_xyz (B, NPOINT, 3) to workspace.
// ---------------------------------------------------------------------------
__global__ __launch_bounds__(1024) void fps_kernel(const float* __restrict__ xyz,
                                                   float* __restrict__ newxyz) {
    const int b = blockIdx.x;
    const int t = threadIdx.x;
    const float* bx = xyz + (size_t)b * N_ * 3;

    float px[4], py[4], pz[4], dist[4];
#pragma unroll
    for (int k = 0; k < 4; ++k) {
        const int p = t + 1024 * k;
        px[k] = bx[p * 3 + 0];
        py[k] = bx[p * 3 + 1];
        pz[k] = bx[p * 3 + 2];
        dist[k] = 1.0e10f;
    }

    __shared__ float cen[3];
    __shared__ float wdist[32];
    __shared__ int   widx[32];
    __shared__ int   sel;

    if (t == 0) { cen[0] = bx[0]; cen[1] = bx[1]; cen[2] = bx[2]; }
    __syncthreads();

    for (int i = 0; i < NPOINT_; ++i) {
        const float cx = cen[0], cy = cen[1], cz = cen[2];
        if (t == 0) {
            float* o = newxyz + ((size_t)b * NPOINT_ + i) * 3;
            o[0] = cx; o[1] = cy; o[2] = cz;
        }

        float val = -1.0f; int best = 0;
#pragma unroll
        for (int k = 0; k < 4; ++k) {
            const float dx = px[k] - cx, dy = py[k] - cy, dz = pz[k] - cz;
            const float d = dx * dx + dy * dy + dz * dz;
            dist[k] = fminf(dist[k], d);
            if (dist[k] > val) { val = dist[k]; best = t + 1024 * k; }
        }
        // wave32 argmax, tie-break to smallest index (matches jnp.argmax)
#pragma unroll
        for (int off = 16; off > 0; off >>= 1) {
            const float ov = __shfl_xor(val, off, 32);
            const int   oi = __shfl_xor(best, off, 32);
            if (ov > val || (ov == val && oi < best)) { val = ov; best = oi; }
        }
        if ((t & 31) == 0) { wdist[t >> 5] = val; widx[t >> 5] = best; }
        __syncthreads();
        if (t < 32) {
            val = wdist[t]; best = widx[t];
#pragma unroll
            for (int off = 16; off > 0; off >>= 1) {
                const float ov = __shfl_xor(val, off, 32);
                const int   oi = __shfl_xor(best, off, 32);
                if (ov > val || (ov == val && oi < best)) { val = ov; best = oi; }
            }
            if (t == 0) sel = best;
        }
        __syncthreads();
        const int j = sel;
        if ((j & 1023) == t) {                 // owner thread updates centroid
            const int k = j >> 10;             // 0..3, select without scratch
            float ax = px[0], ay = py[0], az = pz[0];
            if (k == 1) { ax = px[1]; ay = py[1]; az = pz[1]; }
            if (k == 2) { ax = px[2]; ay = py[2]; az = pz[2]; }
            if (k == 3) { ax = px[3]; ay = py[3]; az = pz[3]; }
            cen[0] = ax; cen[1] = ay; cen[2] = az;
        }
        __syncthreads();
    }
}

// ---------------------------------------------------------------------------
// Kernel B: radius masks via V_WMMA_F32_16X16X4_F32.
// One wave handles 16 centroids; loops over 256 tiles of 16 points.
// D = A(16x4) * B(4x16) + C with A row = (-2sx,-2sy,-2sz,1),
// B col = (px,py,pz,|p|^2), C = |s|^2  =>  D = squared distance tile.
// Ballot per D-VGPR -> 16-bit masks, packed into u32 words (2 point tiles).
// maskbuf layout: [(b*128 + w) * 1024 + s]  (coalesced stores).
// ---------------------------------------------------------------------------
__global__ __launch_bounds__(128) void ballmask_kernel(const float* __restrict__ xyz,
                                                       const float* __restrict__ newxyz,
                                                       unsigned* __restrict__ maskbuf) {
    const int lane = threadIdx.x & 31;
    const int gw = blockIdx.x * 4 + (threadIdx.x >> 5);   // 0..511
    const int b  = gw >> 6;                                // 64 centroid tiles / batch
    const int ct = gw & 63;
    const int r  = lane & 15;                              // row (centroid) / col (point)

    // A matrix (16x4 f32): VGPR0 = K0 | K2, VGPR1 = K1 | K3 (lo | hi lanes)
    const float* s = newxyz + ((size_t)b * NPOINT_ + ct * 16 + r) * 3;
    const float sx = s[0], sy = s[1], sz = s[2];
    const float s2 = sx * sx + sy * sy + sz * sz;
    v2f A;
    A.x = (lane < 16) ? (-2.0f * sx) : (-2.0f * sz);
    A.y = (lane < 16) ? (-2.0f * sy) : 1.0f;

    // C matrix: VGPR v: lanes0-15 = s2[row v], lanes16-31 = s2[row v+8]
    v8f C;
#pragma unroll
    for (int v = 0; v < 8; ++v)
        C[v] = __shfl(s2, (lane < 16) ? v : (v + 8), 32);

    const float* bx = xyz + (size_t)b * N_ * 3;
    for (int w = 0; w < 128; ++w) {
        unsigned word = 0;
#pragma unroll
        for (int half = 0; half < 2; ++half) {
            const int n0 = (w * 2 + half) * 16 + r;
            const float* p = bx + n0 * 3;
            const float qx = p[0], qy = p[1], qz = p[2];
            const float d2 = qx * qx + qy * qy + qz * qz;
            v2f Bm;
            Bm.x = (lane < 16) ? qx : qz;   // K0 | K2
            Bm.y = (lane < 16) ? qy : d2;   // K1 | K3
            v8f D = __builtin_amdgcn_wmma_f32_16x16x4_f32(
                false, A, false, Bm, (short)0, C, false, false);
            unsigned my16 = 0;
#pragma unroll
            for (int v = 0; v < 8; ++v) {
                const unsigned bal = __builtin_amdgcn_ballot_w32(D[v] <= R2_);
                const unsigned h16 = (r < 8) ? (bal & 0xffffu) : (bal >> 16);
                if ((r & 7) == v) my16 = h16;   // lane r owns centroid row r
            }
            word |= my16 << (16 * half);
        }
        if (lane < 16)
            maskbuf[((size_t)(b * 128 + w)) * NPOINT_ + ct * 16 + r] = word;
    }
}

// ---------------------------------------------------------------------------
// Kernel C: per-centroid first-32-in-radius selection + gather + coalesced
// writeback. One wave (32 lanes == NSAMPLE) per centroid; 4 waves per block.
// Lane k takes the (k+1)-th set bit of the mask stream (index order), pads
// with slot 0 (centroid itself is always in-radius -> matches reference pad).
// Tile (32 x 67 floats) staged in LDS, written back as contiguous float4s.
// ---------------------------------------------------------------------------
__global__ __launch_bounds__(128) void group_kernel(const float* __restrict__ xyz,
                                                    const float* __restrict__ points,
                                                    const float* __restrict__ newxyz,
                                                    const unsigned* __restrict__ maskbuf,
                                                    float* __restrict__ out) {
    __shared__ float tile[4][NSAMPLE_ * OCH_];   // 4 * 2144 floats = 34304 B
    const int lane = threadIdx.x & 31;
    const int wv   = threadIdx.x >> 5;
    const int sg   = blockIdx.x * 4 + wv;        // 0..8191
    const int b    = sg >> 10;
    const int s    = sg & 1023;

    int cnt = 0;
    int myidx = -1;
    for (int w = 0; w < 128 && cnt < NSAMPLE_; ++w) {
        const unsigned word = maskbuf[((size_t)(b * 128 + w)) * NPOINT_ + s];
        const int pc = __popc(word);
        if (myidx < 0 && lane >= cnt && lane < cnt + pc) {
            unsigned tb = word;
            const int k = lane - cnt;
            for (int i = 0; i < k; ++i) tb &= tb - 1;   // drop k lowest set bits
            myidx = w * 32 + (__ffs(tb) - 1);
        }
        cnt += pc;
    }
    int first = __shfl(myidx, 0, 32);
    if (first < 0) first = 0;          // unreachable (self is in-radius), safety
    if (myidx < 0) myidx = first;      // pad with slot-0 index

    const float* c = newxyz + ((size_t)b * NPOINT_ + s) * 3;
    const float cx = c[0], cy = c[1], cz = c[2];
    const float* g = xyz + ((size_t)b * N_ + myidx) * 3;
    float* tw = &tile[wv][lane * OCH_];
    tw[0] = g[0] - cx; tw[1] = g[1] - cy; tw[2] = g[2] - cz;

    const float4* pr = (const float4*)(points + ((size_t)b * N_ + myidx) * CCH_);
#pragma unroll
    for (int q = 0; q < CCH_ / 4; ++q) {
        const float4 vv = pr[q];
        tw[3 + q * 4 + 0] = vv.x;
        tw[3 + q * 4 + 1] = vv.y;
        tw[3 + q * 4 + 2] = vv.z;
        tw[3 + q * 4 + 3] = vv.w;
    }
    // coalesced contiguous writeback: 536 float4 per wave tile
    float4* ov = (float4*)(out + (size_t)sg * (NSAMPLE_ * OCH_));
    const float4* tv = (const float4*)&tile[wv][0];
    for (int i = lane; i < (NSAMPLE_ * OCH_) / 4; i += 32) ov[i] = tv[i];
}

// ---------------------------------------------------------------------------
// Workspace layout: [ new_xyz: 8*1024*3 f32 | maskbuf: 8*128*1024 u32 ]
// Total = 98,304 + 4,194,304 = 4,292,608 bytes.
// ---------------------------------------------------------------------------
extern "C" void kernel_launch(void* const* d_in, const int* in_sizes, int n_in,
                              void* d_out, int out_size, void* d_ws, size_t ws_size,
                              hipStream_t stream) {
    const float* xyz    = (const float*)d_in[0];   // (8, 4096, 3)
    const float* points = (const float*)d_in[1];   // (8, 4096, 64)
    float* out = (float*)d_out;                    // (8, 1024, 32, 67)

    float*    newxyz  = (float*)d_ws;
    unsigned* maskbuf = (unsigned*)((char*)d_ws + (size_t)B_ * NPOINT_ * 3 * sizeof(float));

    fps_kernel<<<B_, 1024, 0, stream>>>(xyz, newxyz);
    ballmask_kernel<<<128, 128, 0, stream>>>(xyz, newxyz, maskbuf);
    group_kernel<<<2048, 128, 0, stream>>>(xyz, points, newxyz, maskbuf, out);
}